// _GNNBase_84765474554466
// MI455X (gfx1250) — compile-verified
//
#include <hip/hip_runtime.h>

// GraphSAGE 3-layer forward on MI455X (gfx1250, wave32).
//
// Roofline: GEMM total ~16 GFLOP (negligible). Aggregation: 3 x (819 MB L2
// gathers + ~205M f32 atomics) dominates; node feature table (51.2 MB) fits
// the 192 MB L2. So: f32 end-to-end, V_WMMA_F32_16X16X4_F32 for the GEMMs.

#define NNODES 100000
#define NEDGES 1600000
#define DFEAT  128   // K for every layer (D_IN = H1 = H2 = 128)

typedef __attribute__((ext_vector_type(2))) float v2f;
typedef __attribute__((ext_vector_type(8))) float v8f;

// ---------------- utility: zero a float buffer (n % 4 == 0) ----------------
__global__ void zero_f32(float* __restrict__ p, int n) {
  int i = (blockIdx.x * blockDim.x + threadIdx.x) * 4;
  if (i < n) {
    float4 z = {0.f, 0.f, 0.f, 0.f};
    *(float4*)(p + i) = z;
  }
}

// ---------------- degree: one atomic add per edge ----------------
__global__ void deg_accum(const int* __restrict__ dst, float* __restrict__ deg, int E) {
  int e = blockIdx.x * blockDim.x + threadIdx.x;
  if (e < E) atomicAdd(&deg[dst[e]], 1.0f);
}

// ---------------- scatter-add: one wave32 per edge, 128 floats/row --------
// lane reads float4 of x[src] (global_load_b128, L2-resident) and does 4
// global_atomic_add_f32 into msg[dst].
__global__ void scatter128(const float* __restrict__ x, const int* __restrict__ src,
                           const int* __restrict__ dst, float* __restrict__ msg, int E) {
  int tid  = blockIdx.x * blockDim.x + threadIdx.x;
  int e    = tid >> 5;
  int lane = tid & 31;
  if (e >= E) return;
  int s = src[e];
  int d = dst[e];
  const float4 v = *(const float4*)(x + (size_t)s * DFEAT + lane * 4);
  float* mp = msg + (size_t)d * DFEAT + lane * 4;
  atomicAdd(mp + 0, v.x);
  atomicAdd(mp + 1, v.y);
  atomicAdd(mp + 2, v.z);
  atomicAdd(mp + 3, v.w);
}

// ---------------- fused SAGE layer GEMM (f32 WMMA) ----------------
// out[row, col] = act( (msg[row,:]/clamp(deg,1)) @ Wl + xin[row,:] @ Wr + b )
// One wave -> 16x16 output tile via V_WMMA_F32_16X16X4_F32, K-loop step 4.
// Block = 256 threads = 8 waves: 16 rows x up to 128 cols per block.
//
// Fragment layouts (ISA 7.12.2, wave32):
//   A (16x4 f32): lanes 0-15 hold M=lane, K = k0+{0,1}; lanes 16-31 hold
//                 M=lane-16, K = k0+{2,3}  (VGPR0=first K, VGPR1=second K)
//   B (4x16 f32): mirrored: lane half selects K pair, lane&15 selects N col
//   C/D (16x16):  VGPR r: lanes 0-15 -> (M=r, N=lane); lanes 16-31 -> (M=r+8)
__global__ void sage_gemm_wmma(const float* __restrict__ msg, const float* __restrict__ deg,
                               const float* __restrict__ xin, const float* __restrict__ Wl,
                               const float* __restrict__ Wr, const float* __restrict__ bias,
                               float* __restrict__ out, int K, int H, int do_relu) {
  const int wave = threadIdx.x >> 5;
  const int lane = threadIdx.x & 31;
  const int colBase = wave << 4;
  if (colBase >= H) return;                 // whole wave exits (H=64 case)

  const int rowBase = blockIdx.x << 4;      // N % 16 == 0, no row guard needed
  const int m  = lane & 15;
  const int hf = lane >> 4;                 // lane-half selects K pair
  const int row = rowBase + m;
  const int col = colBase + m;

  const float rdeg = 1.0f / fmaxf(deg[row], 1.0f);
  const float* __restrict__ am = msg + (size_t)row * K;
  const float* __restrict__ ax = xin + (size_t)row * K;

  v8f c = {};
  for (int k0 = 0; k0 < K; k0 += 4) {
    const int kb = k0 + (hf << 1);
    v2f a, b;
    a.x = am[kb] * rdeg;
    a.y = am[kb + 1] * rdeg;
    b.x = Wl[(size_t)kb * H + col];
    b.y = Wl[(size_t)(kb + 1) * H + col];
    c = __builtin_amdgcn_wmma_f32_16x16x4_f32(false, a, false, b, (short)0, c, false, false);

    v2f a2, b2;
    a2.x = ax[kb];
    a2.y = ax[kb + 1];
    b2.x = Wr[(size_t)kb * H + col];
    b2.y = Wr[(size_t)(kb + 1) * H + col];
    c = __builtin_amdgcn_wmma_f32_16x16x4_f32(false, a2, false, b2, (short)0, c, false, false);
  }

  const float bv = bias[col];
#pragma unroll
  for (int r = 0; r < 8; ++r) {
    int orow = rowBase + r + (hf << 3);
    float v = c[r] + bv;
    if (do_relu) v = fmaxf(v, 0.0f);
    out[(size_t)orow * H + col] = v;
  }
}

// ---------------- host-side launch ----------------
extern "C" void kernel_launch(void* const* d_in, const int* in_sizes, int n_in,
                              void* d_out, int out_size, void* d_ws, size_t ws_size,
                              hipStream_t stream) {
  const float* x    = (const float*)d_in[0];
  const int*   edge = (const int*)d_in[1];
  const int*   src  = edge;            // edge_index[0]
  const int*   dst  = edge + NEDGES;   // edge_index[1]
  const float* W1l = (const float*)d_in[2];
  const float* W1r = (const float*)d_in[3];
  const float* b1  = (const float*)d_in[4];
  const float* W2l = (const float*)d_in[5];
  const float* W2r = (const float*)d_in[6];
  const float* b2  = (const float*)d_in[7];
  const float* W3l = (const float*)d_in[8];
  const float* W3r = (const float*)d_in[9];
  const float* b3  = (const float*)d_in[10];

  // Workspace layout (all offsets 16B-aligned):
  float* deg = (float*)d_ws;                        // NNODES
  float* msg = deg + NNODES;                        // NNODES*128
  float* h1  = msg + (size_t)NNODES * DFEAT;        // NNODES*128
  float* h2  = h1 + (size_t)NNODES * DFEAT;         // NNODES*128
  float* outp = (float*)d_out;                      // NNODES*64

  const int TB = 256;
  const int nMsg = NNODES * DFEAT;
  dim3 zeroMsgGrid((nMsg / 4 + TB - 1) / TB);
  dim3 zeroDegGrid((NNODES / 4 + TB - 1) / TB);
  dim3 degGrid((NEDGES + TB - 1) / TB);
  dim3 scatGrid(NEDGES * 32 / TB);                  // one wave per edge
  dim3 gemmGrid(NNODES / 16);

  // degree (shared by all layers)
  zero_f32<<<zeroDegGrid, TB, 0, stream>>>(deg, NNODES);
  deg_accum<<<degGrid, TB, 0, stream>>>(dst, deg, NEDGES);

  // ---- layer 1: x -> h1 (H=128, relu) ----
  zero_f32<<<zeroMsgGrid, TB, 0, stream>>>(msg, nMsg);
  scatter128<<<scatGrid, TB, 0, stream>>>(x, src, dst, msg, NEDGES);
  sage_gemm_wmma<<<gemmGrid, TB, 0, stream>>>(msg, deg, x, W1l, W1r, b1, h1,
                                              DFEAT, 128, 1);

  // ---- layer 2: h1 -> h2 (H=128, relu) ----
  zero_f32<<<zeroMsgGrid, TB, 0, stream>>>(msg, nMsg);
  scatter128<<<scatGrid, TB, 0, stream>>>(h1, src, dst, msg, NEDGES);
  sage_gemm_wmma<<<gemmGrid, TB, 0, stream>>>(msg, deg, h1, W2l, W2r, b2, h2,
                                              DFEAT, 128, 1);

  // ---- layer 3: h2 -> out (H=64, no relu) ----
  zero_f32<<<zeroMsgGrid, TB, 0, stream>>>(msg, nMsg);
  scatter128<<<scatGrid, TB, 0, stream>>>(h2, src, dst, msg, NEDGES);
  sage_gemm_wmma<<<gemmGrid, TB, 0, stream>>>(msg, deg, h2, W3l, W3r, b3, outp,
                                              DFEAT, 64, 0);
}